// DegreeGCNLayer_27642409517680
// MI455X (gfx1250) — compile-verified
//
#include <hip/hip_runtime.h>

typedef __attribute__((ext_vector_type(2))) float v2f;
typedef __attribute__((ext_vector_type(8))) float v8f;

#define FEATS 64  // IN_FEATS == OUT_FEATS == 64

// ---------------------------------------------------------------------------
// Kernel 1: zero the aggregation workspace (float4-wide, grid-stride)
// ---------------------------------------------------------------------------
__global__ void gcn_zero_kernel(float4* __restrict__ p, int n4) {
    int i = blockIdx.x * blockDim.x + threadIdx.x;
    int stride = gridDim.x * blockDim.x;
    float4 z = make_float4(0.f, 0.f, 0.f, 0.f);
    for (; i < n4; i += stride) p[i] = z;
}

// ---------------------------------------------------------------------------
// Kernel 2: edge scatter.  One wave32 per edge (grid-stride over edges).
// Lane l handles feature elements 2l, 2l+1 as a float2 (global_load_b64),
// scales by 1/sqrt(deg[src]), and scatter-adds with native f32 L2 atomics.
// ---------------------------------------------------------------------------
__global__ void gcn_edge_scatter_kernel(const float* __restrict__ feat,
                                        const float* __restrict__ degree,
                                        const int* __restrict__ src,
                                        const int* __restrict__ dst,
                                        float* __restrict__ agg,
                                        int n_edges) {
    int wave   = (blockIdx.x * blockDim.x + threadIdx.x) >> 5;
    int lane   = threadIdx.x & 31;
    int nwaves = (gridDim.x * blockDim.x) >> 5;

    for (int e = wave; e < n_edges; e += nwaves) {
        int s = src[e];
        int d = dst[e];
        float w = __frsqrt_rn(degree[s]);
        float2 v = *(const float2*)(feat + (size_t)s * FEATS + lane * 2);
        float* outp = agg + (size_t)d * FEATS + lane * 2;
        // native global_atomic_add_f32 (no return) -> L2 near-atomics
        unsafeAtomicAdd(outp + 0, v.x * w);
        unsafeAtomicAdd(outp + 1, v.y * w);
    }
}

// ---------------------------------------------------------------------------
// Kernel 3: node apply  h = (agg * rsqrt(deg)) @ W + b   via WMMA f32 16x16x4.
// One wave computes a 16(M) x 64(N) output stripe:
//   A frag (lane l, group g=l>>4, m=l&15): contiguous float2 of scaled agg row
//   B frag: W[k0+2g][ntile*16 + (l&15)], W[k0+2g+1][...]
//   C/D:    VGPR r -> row mtile*16 + r + 8*g, col = l&15 within ntile
// ---------------------------------------------------------------------------
__global__ void gcn_node_apply_wmma_kernel(const float* __restrict__ agg,
                                           const float* __restrict__ degree,
                                           const float* __restrict__ Wm,
                                           const float* __restrict__ bias,
                                           float* __restrict__ out,
                                           int n_nodes, int n_mtiles) {
    int wave  = blockIdx.x * (blockDim.x >> 5) + (threadIdx.x >> 5);
    if (wave >= n_mtiles) return;                 // uniform per-wave exit
    int lane  = threadIdx.x & 31;
    int n     = lane & 15;                        // column within an N-tile / row within M-tile
    int g     = lane >> 4;                        // lane half-group

    int m      = wave * 16 + n;                   // this lane's A-matrix row
    int mclamp = m < n_nodes ? m : n_nodes - 1;   // clamp loads, keep EXEC all-1s
    float dsc  = __frsqrt_rn(degree[mclamp]);
    const float* arow = agg + (size_t)mclamp * FEATS;

    v8f acc0 = {}, acc1 = {}, acc2 = {}, acc3 = {};

    for (int k0 = 0; k0 < FEATS; k0 += 4) {
        int ka = k0 + 2 * g;
        float2 av = *(const float2*)(arow + ka);
        v2f a;
        a.x = av.x * dsc;
        a.y = av.y * dsc;

        const float* wr0 = Wm + (size_t)ka * FEATS;        // W row ka
        const float* wr1 = wr0 + FEATS;                    // W row ka+1
        v2f b0; b0.x = wr0[n +  0]; b0.y = wr1[n +  0];
        v2f b1; b1.x = wr0[n + 16]; b1.y = wr1[n + 16];
        v2f b2; b2.x = wr0[n + 32]; b2.y = wr1[n + 32];
        v2f b3; b3.x = wr0[n + 48]; b3.y = wr1[n + 48];

        acc0 = __builtin_amdgcn_wmma_f32_16x16x4_f32(false, a, false, b0, (short)0, acc0, false, false);
        acc1 = __builtin_amdgcn_wmma_f32_16x16x4_f32(false, a, false, b1, (short)0, acc1, false, false);
        acc2 = __builtin_amdgcn_wmma_f32_16x16x4_f32(false, a, false, b2, (short)0, acc2, false, false);
        acc3 = __builtin_amdgcn_wmma_f32_16x16x4_f32(false, a, false, b3, (short)0, acc3, false, false);
    }

    float bb0 = bias[n +  0];
    float bb1 = bias[n + 16];
    float bb2 = bias[n + 32];
    float bb3 = bias[n + 48];

#pragma unroll
    for (int r = 0; r < 8; ++r) {
        int mrow = wave * 16 + r + 8 * g;         // C/D layout: VGPR r -> M = r (+8 for hi half)
        if (mrow < n_nodes) {
            float* orow = out + (size_t)mrow * FEATS + n;
            orow[ 0] = acc0[r] + bb0;
            orow[16] = acc1[r] + bb1;
            orow[32] = acc2[r] + bb2;
            orow[48] = acc3[r] + bb3;
        }
    }
}

// ---------------------------------------------------------------------------
// Launcher.  Inputs (setup_inputs order):
//   0: feature [N,64] f32   1: degree [N] f32   2: src [E] i32
//   3: dst [E] i32          4: W [64,64] f32    5: b [64] f32
// d_out: [N,64] f32.  d_ws: agg accumulator (N*64 f32 = 19.2 MB).
// ---------------------------------------------------------------------------
extern "C" void kernel_launch(void* const* d_in, const int* in_sizes, int n_in,
                              void* d_out, int out_size, void* d_ws, size_t ws_size,
                              hipStream_t stream) {
    const float* feature = (const float*)d_in[0];
    const float* degree  = (const float*)d_in[1];
    const int*   src     = (const int*)d_in[2];
    const int*   dst     = (const int*)d_in[3];
    const float* Wm      = (const float*)d_in[4];
    const float* bias    = (const float*)d_in[5];
    float* out = (float*)d_out;
    float* agg = (float*)d_ws;

    int n_nodes = in_sizes[1];
    int n_edges = in_sizes[2];

    // 1) zero agg
    int n4 = (n_nodes * FEATS) / 4;
    int zb = (n4 + 255) / 256;
    gcn_zero_kernel<<<zb, 256, 0, stream>>>((float4*)agg, n4);

    // 2) edge scatter: 8192 waves grid-striding 1.2M edges
    gcn_edge_scatter_kernel<<<1024, 256, 0, stream>>>(feature, degree, src, dst,
                                                      agg, n_edges);

    // 3) node apply with WMMA: one wave per 16-row M-tile, 8 waves/block
    int n_mtiles = (n_nodes + 15) / 16;
    int blocks = (n_mtiles + 7) / 8;
    gcn_node_apply_wmma_kernel<<<blocks, 256, 0, stream>>>(agg, degree, Wm, bias,
                                                           out, n_nodes, n_mtiles);
}